// O3TensorProduct_64793876627489
// MI455X (gfx1250) — compile-verified
//
#include <hip/hip_runtime.h>

typedef __attribute__((ext_vector_type(2))) float v2f;
typedef __attribute__((ext_vector_type(8))) float v8f;

#define WMMA_F32(A, B, C) \
  __builtin_amdgcn_wmma_f32_16x16x4_f32(false, (A), false, (B), (short)0, (C), false, false)

struct F3 { float a, b, c; };   // 12B, 4B-aligned coalesced o1 store

#define TILE_M 16
#define KDIM 128
#define NSTEPS 32               // K=128 / 4 per WMMA
#define INV_SQRT3 0.57735026918962576451f
#define MAX_WG 1250

// LDS layouts (all conflict-free b64 reads in the hot loop):
//  sW[g] (B-frag order): element (k,n) at (((t*8+cb)*2+hh)*16+nn)*2+vv
//        with k=4t+2*hh+vv, n=16*cb+nn.  Read: lane half picks hh -> halves
//        land 32 banks apart; v pair contiguous -> ds_load_b64.
//  sA[g] (A-frag order, XOR swizzle): element (m,k) at
//        ((t*16 + (m ^ (t&15)))*4 + 2*hh + vv), k=4t+2*hh+vv.
//        Read banks: 4*m' + 2*half + v -> all 32 lanes distinct banks.

__global__ __launch_bounds__(256, 1)
void o3tp_kernel(const float* __restrict__ x, const float* __restrict__ y,
                 const float* __restrict__ w000, const float* __restrict__ w110,
                 const float* __restrict__ w011, const float* __restrict__ w101,
                 const float* __restrict__ bias, float* __restrict__ out,
                 int n, int ntiles)
{
  __shared__ float sW[4 * KDIM * KDIM];     // 256 KB: w000,w110,w011,w101
  __shared__ float sA[6 * TILE_M * KDIM];   //  48 KB: 6 derived A tiles

  const int tid  = threadIdx.x;
  const int lane = tid & 31;
  const int wave = tid >> 5;                 // 0..7 -> 16-col output block
  const int half = lane >> 4;                // lane-half (0/1)
  const int m16  = lane & 15;

  // ---- Stage all four weight matrices into LDS (once per block) ----
  {
    const float* wsrc[4] = { w000, w110, w011, w101 };
    for (int g = 0; g < 4; ++g) {
      const float4* src = (const float4*)wsrc[g];
      float* dst = sW + g * (KDIM * KDIM);
      for (int i4 = tid; i4 < (KDIM * KDIM) / 4; i4 += 256) {
        float4 wv = src[i4];                 // coalesced global b128
        int k  = i4 >> 5;                    // row (contraction index u)
        int n0 = (i4 & 31) * 4;              // first of 4 columns
        int t = k >> 2, hh = (k >> 1) & 1, vv = k & 1;
        float el[4] = { wv.x, wv.y, wv.z, wv.w };
#pragma unroll
        for (int i = 0; i < 4; ++i) {
          int nn = n0 + i;
          int idx = (((t * 8 + (nn >> 4)) * 2 + hh) * 16 + (nn & 15)) * 2 + vv;
          dst[idx] = el[i];
        }
      }
    }
  }
  __syncthreads();

  // ---- Persistent loop over 16-sample tiles ----
  for (int tile = blockIdx.x; tile < ntiles; tile += gridDim.x) {
    const int s0 = tile * TILE_M;

    // Prefetch next tile's x rows (global_prefetch_b8)
    int ptile = tile + gridDim.x;
    if (ptile < ntiles) {
      const char* px = (const char*)(x + (size_t)ptile * TILE_M * 512);
      __builtin_prefetch(px + tid * 128, 0, 1);
    }

    // Build the 6 derived A matrices into LDS (coalesced x reads)
#pragma unroll
    for (int j = 0; j < 8; ++j) {
      int p = tid + j * 256;                 // 0..2047 over (s,u)
      int s = p >> 7;                        // sample in tile
      int u = p & 127;                       // contraction index
      const float* xr = x + (size_t)(s0 + s) * 512;
      const float* yr = y + (size_t)(s0 + s) * 4;
      float x0v = xr[u];
      float xa  = xr[128 + 3 * u + 0];
      float xb  = xr[128 + 3 * u + 1];
      float xc  = xr[128 + 3 * u + 2];
      float y0v = yr[0], y1a = yr[1], y1b = yr[2], y1c = yr[3];
      int t = u >> 2, r = u & 3;
      int base = (t * 16 + (s ^ (t & 15))) * 4 + r;   // A-frag + swizzle
      float f = INV_SQRT3 * y0v;
      sA[0 * 2048 + base] = x0v * y0v;                            // -> w000
      sA[1 * 2048 + base] = INV_SQRT3 * (xa*y1a + xb*y1b + xc*y1c); // -> w110
      sA[2 * 2048 + base] = x0v;                                  // -> w011
      sA[3 * 2048 + base] = f * xa;                               // -> w101
      sA[4 * 2048 + base] = f * xb;
      sA[5 * 2048 + base] = f * xc;
    }
    __syncthreads();

    // ---- 6 fused GEMMs: 16 samples x 16 cols per wave, K=128 ----
    v8f acc0  = {0,0,0,0,0,0,0,0};   // o0 (w000 + w110 chained)
    v8f accS  = {0,0,0,0,0,0,0,0};   // x0 @ w011 (scaled per-k in epilogue)
    v8f acc30 = {0,0,0,0,0,0,0,0};   // (is3*y0*x1_k) @ w101
    v8f acc31 = {0,0,0,0,0,0,0,0};
    v8f acc32 = {0,0,0,0,0,0,0,0};

#pragma unroll 4
    for (int t = 0; t < NSTEPS; ++t) {
      const float* pa = sA + (t * 16 + (m16 ^ (t & 15))) * 4 + 2 * half;
      v2f a0  = *(const v2f*)(pa);
      v2f a1  = *(const v2f*)(pa + 2048);
      v2f a2  = *(const v2f*)(pa + 4096);
      v2f a30 = *(const v2f*)(pa + 6144);
      v2f a31 = *(const v2f*)(pa + 8192);
      v2f a32 = *(const v2f*)(pa + 10240);

      const float* pb = sW + (((t * 8 + wave) * 2 + half) * 16 + m16) * 2;
      v2f b000 = *(const v2f*)(pb);
      v2f b110 = *(const v2f*)(pb + 16384);
      v2f b011 = *(const v2f*)(pb + 32768);
      v2f b101 = *(const v2f*)(pb + 49152);

      acc0  = WMMA_F32(a0,  b000, acc0);
      acc0  = WMMA_F32(a1,  b110, acc0);
      accS  = WMMA_F32(a2,  b011, accS);
      acc30 = WMMA_F32(a30, b101, acc30);
      acc31 = WMMA_F32(a31, b101, acc31);
      acc32 = WMMA_F32(a32, b101, acc32);
    }

    // ---- Epilogue: C/D frag: lane=N col, vgpr v -> sample v + 8*half ----
    const int ncol = wave * 16 + m16;
    const float bv = bias[ncol];
#pragma unroll
    for (int v = 0; v < 8; ++v) {
      int srow = s0 + half * 8 + v;
      if (srow < n) {
        float* orow = out + (size_t)srow * 512;
        const float* yr = y + (size_t)srow * 4;
        orow[ncol] = acc0[v] + bv;
        float z2 = INV_SQRT3 * accS[v];
        F3 o1 = { z2 * yr[1] + acc30[v],
                  z2 * yr[2] + acc31[v],
                  z2 * yr[3] + acc32[v] };
        *(F3*)(orow + 128 + ncol * 3) = o1;   // coalesced b96
      }
    }
    __syncthreads();   // protect sA before next tile's rebuild
  }
}

extern "C" void kernel_launch(void* const* d_in, const int* in_sizes, int n_in,
                              void* d_out, int out_size, void* d_ws, size_t ws_size,
                              hipStream_t stream) {
  const float* x    = (const float*)d_in[0];
  const float* y    = (const float*)d_in[1];
  const float* w000 = (const float*)d_in[2];
  const float* w110 = (const float*)d_in[3];
  const float* w011 = (const float*)d_in[4];
  const float* w101 = (const float*)d_in[5];
  const float* b    = (const float*)d_in[6];
  float* out = (float*)d_out;

  int n      = in_sizes[0] / 512;
  int ntiles = (n + TILE_M - 1) / TILE_M;
  int grid   = ntiles < MAX_WG ? ntiles : MAX_WG;

  o3tp_kernel<<<dim3(grid), dim3(256), 0, stream>>>(
      x, y, w000, w110, w011, w101, b, out, n, ntiles);
}